// GCN_18691697672400
// MI455X (gfx1250) — compile-verified
//
#include <hip/hip_runtime.h>
#include <hip/hip_bf16.h>

// GCN forward on MI455X (gfx1250, wave32).
// Roofline: GEMMs are ~7.4 GFLOP (trivial); the edge gather/scatter phase moves
// ~1.9GB but H (25.6MB) and pre (25.6MB) are L2-resident (192MB L2), so the
// scatter is L2-atomic bound. GEMM uses V_WMMA_F32_16X16X4_F32 (full fp32,
// matching the fp32 reference; no precision drop needed on a memory-bound op).

typedef __attribute__((ext_vector_type(2))) float v2f;
typedef __attribute__((ext_vector_type(8))) float v8f;

#define NNODES 50000
#define BN_EPS 1e-5f

// ---------------------------------------------------------------------------
// GEMM: C[M,N] = A[M,256] * W[256,N], fp32 via WMMA 16x16x4.
// One wave per 16x16 tile. A-fragment layout (16x4 f32):
//   lanes 0-15 : VGPR0=K0, VGPR1=K1 for row M=lane
//   lanes 16-31: VGPR0=K2, VGPR1=K3 for row M=lane-16
// B-fragment (4x16) mirrored: lanes0-15 hold rows K0/K1, lanes16-31 K2/K3.
// C/D layout: VGPR i -> M = i + 8*(lane>=16), N = lane&15.
// ---------------------------------------------------------------------------
template <int N>
__global__ void __launch_bounds__(256)
wmma_gemm_kernel(const float* __restrict__ A, const float* __restrict__ W,
                 float* __restrict__ C, int M, int numTiles) {
    constexpr int K = 256;
    constexpr int NT = N / 16;
    const int wave = threadIdx.x >> 5;
    const int lane = threadIdx.x & 31;
    const int tile = blockIdx.x * (blockDim.x >> 5) + wave;
    if (tile >= numTiles) return;
    const int mt = tile / NT;
    const int nt = tile - mt * NT;
    const int half = lane >> 4;   // 0: K0/K1 pair, 1: K2/K3 pair
    const int l    = lane & 15;

    const float* __restrict__ arow = A + (size_t)(mt * 16 + l) * K;
    const float* __restrict__ bcol = W + nt * 16 + l;

    v8f acc = {};
#pragma unroll 8
    for (int k0 = 0; k0 < K; k0 += 4) {
        const int ka = k0 + 2 * half;
        // A: two consecutive fp32 (8B aligned since ka is even)
        v2f a = *(const v2f*)(arow + ka);
        // B: coalesced across 16 lanes, two K-rows
        v2f b;
        b.x = bcol[(size_t)(ka + 0) * N];
        b.y = bcol[(size_t)(ka + 1) * N];
        acc = __builtin_amdgcn_wmma_f32_16x16x4_f32(
            /*neg_a=*/false, a, /*neg_b=*/false, b,
            /*c_mod=*/(short)0, acc, /*reuse_a=*/false, /*reuse_b=*/false);
    }

    float* __restrict__ crow = C + (size_t)(mt * 16 + 8 * half) * N + nt * 16 + l;
#pragma unroll
    for (int i = 0; i < 8; ++i) crow[(size_t)i * N] = acc[i];
}

// ---------------------------------------------------------------------------
// Edge scatter: out[dst[e]] += w[e] * H[src[e]]  (fp32 atomics resolve in L2;
// H and out are both L2-resident). One wave per edge; lane handles 4 floats
// (F=128) or 1 float (F=32); gathers are fully coalesced 512B/128B per wave.
// ---------------------------------------------------------------------------
__global__ void __launch_bounds__(256)
scatter128_kernel(const float* __restrict__ H, const int* __restrict__ src,
                  const int* __restrict__ dst, const float* __restrict__ ew,
                  float* __restrict__ out, int E) {
    const int e = blockIdx.x * (blockDim.x >> 5) + (threadIdx.x >> 5);
    if (e >= E) return;
    const int lane = threadIdx.x & 31;
    const int s = src[e];
    const int d = dst[e];
    const float wgt = ew[e];
    const float4 v = ((const float4*)(H + (size_t)s * 128))[lane];
    float* op = out + (size_t)d * 128 + lane * 4;
    atomicAdd(op + 0, wgt * v.x);
    atomicAdd(op + 1, wgt * v.y);
    atomicAdd(op + 2, wgt * v.z);
    atomicAdd(op + 3, wgt * v.w);
}

__global__ void __launch_bounds__(256)
scatter32_kernel(const float* __restrict__ H, const int* __restrict__ src,
                 const int* __restrict__ dst, const float* __restrict__ ew,
                 float* __restrict__ out, int E) {
    const int e = blockIdx.x * (blockDim.x >> 5) + (threadIdx.x >> 5);
    if (e >= E) return;
    const int lane = threadIdx.x & 31;
    const int s = src[e];
    const int d = dst[e];
    const float wgt = ew[e];
    const float v = H[(size_t)s * 32 + lane];
    atomicAdd(out + (size_t)d * 32 + lane, wgt * v);
}

// ---------------------------------------------------------------------------
// BatchNorm (training stats): column sums / sumsq over [M,128].
// ---------------------------------------------------------------------------
#define BN_ROWS 512
__global__ void __launch_bounds__(256)
bn_stats_kernel(const float* __restrict__ pre, float* __restrict__ stats, int M) {
    const int col = threadIdx.x & 127;
    const int rh  = threadIdx.x >> 7;        // 0..1
    const int rend0 = (blockIdx.x + 1) * BN_ROWS;
    const int rend  = rend0 < M ? rend0 : M;
    float s = 0.f, ss = 0.f;
    for (int r = blockIdx.x * BN_ROWS + rh; r < rend; r += 2) {
        const float v = pre[(size_t)r * 128 + col];
        s += v;
        ss += v * v;
    }
    atomicAdd(&stats[col], s);
    atomicAdd(&stats[128 + col], ss);
}

__global__ void __launch_bounds__(128)
bn_finalize_kernel(const float* __restrict__ stats, const float* __restrict__ gamma,
                   const float* __restrict__ beta, float* __restrict__ scsh, float invN) {
    const int c = threadIdx.x;               // 128 threads
    const float mu  = stats[c] * invN;
    const float var = stats[128 + c] * invN - mu * mu;
    const float sc  = gamma[c] * rsqrtf(var + BN_EPS);
    scsh[c]       = sc;
    scsh[128 + c] = beta[c] - mu * sc;
}

// relu(bn(pre)) || pre  -> xout [M,256]
__global__ void __launch_bounds__(256)
bn_apply_concat_kernel(const float* __restrict__ pre, const float* __restrict__ scsh,
                       float* __restrict__ xout, int M) {
    const int idx = blockIdx.x * blockDim.x + threadIdx.x;
    if (idx >= M * 128) return;
    const int r = idx >> 7;
    const int c = idx & 127;
    const float v = pre[idx];
    const float n = fmaf(v, scsh[c], scsh[128 + c]);
    xout[(size_t)r * 256 + c]       = fmaxf(n, 0.f);
    xout[(size_t)r * 256 + 128 + c] = v;
}

// ---------------------------------------------------------------------------
extern "C" void kernel_launch(void* const* d_in, const int* in_sizes, int n_in,
                              void* d_out, int out_size, void* d_ws, size_t ws_size,
                              hipStream_t stream) {
    const float* x   = (const float*)d_in[0];
    const int*   src = (const int*)d_in[1];
    const int*   dst = (const int*)d_in[2];
    const float* ew  = (const float*)d_in[3];
    const float* W0  = (const float*)d_in[4];
    const float* W1  = (const float*)d_in[5];
    const float* W2  = (const float*)d_in[6];
    const float* g0  = (const float*)d_in[7];
    const float* b0  = (const float*)d_in[8];
    const float* g1  = (const float*)d_in[9];
    const float* b1  = (const float*)d_in[10];

    const int M = NNODES;                 // 50000 (divisible by 16)
    const int E = in_sizes[1];            // 800000
    float* outf = (float*)d_out;

    // workspace layout (~103 MB)
    float* H     = (float*)d_ws;                  // [M,128] (also holds H2 [M,32])
    float* pre   = H + (size_t)M * 128;           // [M,128]
    float* xc    = pre + (size_t)M * 128;         // [M,256] concat buffer
    float* stats = xc + (size_t)M * 256;          // [256]
    float* scsh  = stats + 256;                   // [256]

    const int mTiles   = M / 16;                  // 3125
    const int tiles128 = mTiles * (128 / 16);     // 25000
    const int tiles32  = mTiles * (32 / 16);      // 6250
    const int edgeBlocks = (E + 7) / 8;
    const int bnBlocks   = (M + BN_ROWS - 1) / BN_ROWS;
    const int applyBlocks = (M * 128 + 255) / 256;
    const float invN = 1.0f / (float)M;

    // ---------------- layer 0 ----------------
    wmma_gemm_kernel<128><<<(tiles128 + 7) / 8, 256, 0, stream>>>(x, W0, H, M, tiles128);
    hipMemsetAsync(pre, 0, (size_t)M * 128 * sizeof(float), stream);
    scatter128_kernel<<<edgeBlocks, 256, 0, stream>>>(H, src, dst, ew, pre, E);
    hipMemsetAsync(stats, 0, 256 * sizeof(float), stream);
    bn_stats_kernel<<<bnBlocks, 256, 0, stream>>>(pre, stats, M);
    bn_finalize_kernel<<<1, 128, 0, stream>>>(stats, g0, b0, scsh, invN);
    bn_apply_concat_kernel<<<applyBlocks, 256, 0, stream>>>(pre, scsh, xc, M);

    // ---------------- layer 1 ----------------
    wmma_gemm_kernel<128><<<(tiles128 + 7) / 8, 256, 0, stream>>>(xc, W1, H, M, tiles128);
    hipMemsetAsync(pre, 0, (size_t)M * 128 * sizeof(float), stream);
    scatter128_kernel<<<edgeBlocks, 256, 0, stream>>>(H, src, dst, ew, pre, E);
    hipMemsetAsync(stats, 0, 256 * sizeof(float), stream);
    bn_stats_kernel<<<bnBlocks, 256, 0, stream>>>(pre, stats, M);
    bn_finalize_kernel<<<1, 128, 0, stream>>>(stats, g1, b1, scsh, invN);
    bn_apply_concat_kernel<<<applyBlocks, 256, 0, stream>>>(pre, scsh, xc, M);

    // ---------------- output layer ----------------
    wmma_gemm_kernel<32><<<(tiles32 + 7) / 8, 256, 0, stream>>>(xc, W2, H, M, tiles32);
    hipMemsetAsync(outf, 0, (size_t)M * 32 * sizeof(float), stream);
    scatter32_kernel<<<edgeBlocks, 256, 0, stream>>>(H, src, dst, ew, outf, E);
}